// Grouping_27041114096412
// MI455X (gfx1250) — compile-verified
//
#include <hip/hip_runtime.h>
#include <hip/hip_bf16.h>
#include <stdint.h>

typedef __attribute__((ext_vector_type(2))) float v2f;
typedef __attribute__((ext_vector_type(8))) float v8f;

#define NPTS 16384
#define NCTR 4096
#define KNN  32

__device__ __forceinline__ unsigned long long shfl_xor_u64(unsigned long long v, int mask) {
    int lo = __shfl_xor((int)(unsigned)(v & 0xffffffffull), mask, 32);
    int hi = __shfl_xor((int)(unsigned)(v >> 32), mask, 32);
    return ((unsigned long long)(unsigned)hi << 32) | (unsigned)lo;
}

// One bitonic compare-exchange step for element index i against partner i^j,
// stage size k (ascending iff (i & k) == 0).
__device__ __forceinline__ unsigned long long cex(unsigned long long mine, int i, int j, int k) {
    unsigned long long part = shfl_xor_u64(mine, j);
    bool lower   = (i & j) == 0;
    bool asc     = (i & k) == 0;
    bool takeMin = (lower == asc);
    bool pLess   = part < mine;
    return (pLess == takeMin) ? part : mine;
}

__global__ __launch_bounds__(128) void knn_wmma_kernel(const float* __restrict__ x,
                                                       const float* __restrict__ cen,
                                                       int* __restrict__ out) {
    // Per-wave private top-32 candidate lists, [entry][lane] for conflict-free scans.
    __shared__ float        sdist[4][32][32];
    __shared__ unsigned int sidx [4][32][32];

    const int lane  = threadIdx.x & 31;
    const int w     = threadIdx.x >> 5;
    const int tile  = blockIdx.x * 4 + w;      // 0..1023  (B * M/16 tiles)
    const int b     = tile >> 8;               // batch 0..3
    const int cbase = (tile & 255) << 4;       // centroid tile base (16 per wave)

    const int col  = lane & 15;                // C-matrix column = centroid owned by lane
    const int half = lane >> 4;                // C-matrix row half = point sub-range

    // ---- B operand (constant over whole N loop): column cbase+col, K = 2*half+{0,1}
    // B[k][c] = { -2*c0, -2*c1, -2*c2, 1 }
    const float* cp = cen + ((size_t)b * NCTR + (size_t)(cbase + col)) * 3;
    float c0 = cp[0], c1 = cp[1], c2 = cp[2];
    v2f bmat;
    bmat.x = half ? (-2.0f * c2) : (-2.0f * c0);
    bmat.y = half ? 1.0f         : (-2.0f * c1);

    // ---- init candidate lists
    const float INF = __builtin_inff();
    #pragma unroll
    for (int e = 0; e < 32; ++e) {
        sdist[w][e][lane] = INF;
        sidx [w][e][lane] = 0u;
    }
    float curMax  = INF;
    int   curSlot = 0;

    const float* xb = x + (size_t)b * NPTS * 3;

    // ---- main loop: 16 points per WMMA, 1024 iterations
    for (int nb = 0; nb < NPTS; nb += 16) {
        // A operand: row m = lane%16 = point nb+col, A[m] = (px, py, pz, |p|^2)
        const float* pp = xb + (size_t)(nb + col) * 3;
        float px = pp[0], py = pp[1], pz = pp[2];
        float p2 = px * px + py * py + pz * pz;
        v2f a;
        a.x = half ? pz : px;
        a.y = half ? p2 : py;

        v8f acc = {};
        // D[p][c] = |p|^2 - 2 p.c   (per-centroid constant |c|^2 and sqrt are rank-invariant)
        acc = __builtin_amdgcn_wmma_f32_16x16x4_f32(false, a, false, bmat,
                                                    (short)0, acc, false, false);

        #pragma unroll
        for (int v = 0; v < 8; ++v) {
            float d = acc[v];                       // point nb + 8*half + v, centroid col
            if (d < curMax) {
                sdist[w][curSlot][lane] = d;
                sidx [w][curSlot][lane] = (unsigned)(nb + half * 8 + v);
                // rescan for new max / argmax
                float m = sdist[w][0][lane];
                int   s = 0;
                #pragma unroll
                for (int e = 1; e < 32; ++e) {
                    float t = sdist[w][e][lane];
                    if (t > m) { m = t; s = e; }
                }
                curMax = m; curSlot = s;
            }
        }
    }

    __syncthreads();   // waves are independent; barrier just for LDS safety

    // ---- per centroid: merge the lane-pair's 2x32 candidates, bitonic sort 64, emit 32
    for (int c = 0; c < 16; ++c) {
        float    d0 = sdist[w][lane][c];
        unsigned i0 = sidx [w][lane][c];
        float    d1 = sdist[w][lane][c + 16];
        unsigned i1 = sidx [w][lane][c + 16];

        // ordered-uint mapping (distances can be negative after dropping |c|^2)
        unsigned k0 = __float_as_uint(d0); k0 = (k0 & 0x80000000u) ? ~k0 : (k0 | 0x80000000u);
        unsigned k1 = __float_as_uint(d1); k1 = (k1 & 0x80000000u) ? ~k1 : (k1 | 0x80000000u);
        unsigned long long r0 = ((unsigned long long)k0 << 32) | i0;   // element i = lane
        unsigned long long r1 = ((unsigned long long)k1 << 32) | i1;   // element i = lane+32

        // bitonic sort, 64 elements, ascending
        #pragma unroll
        for (int k = 2; k <= 32; k <<= 1) {
            #pragma unroll
            for (int j = k >> 1; j > 0; j >>= 1) {
                r0 = cex(r0, lane,      j, k);
                r1 = cex(r1, lane + 32, j, k);
            }
        }
        {   // k=64, j=32: partner lives in the other register of the same lane
            unsigned long long lo = (r0 < r1) ? r0 : r1;
            unsigned long long hi = (r0 < r1) ? r1 : r0;
            r0 = lo; r1 = hi;
        }
        #pragma unroll
        for (int j = 16; j > 0; j >>= 1) {
            r0 = cex(r0, lane,      j, 64);
            r1 = cex(r1, lane + 32, j, 64);
        }

        // r0 of lane l == rank-l nearest neighbour (idx in low 32 bits)
        out[((size_t)b * NCTR + (size_t)(cbase + c)) * KNN + lane] =
            (int)(unsigned)(r0 & 0xffffffffull);
    }
}

extern "C" void kernel_launch(void* const* d_in, const int* in_sizes, int n_in,
                              void* d_out, int out_size, void* d_ws, size_t ws_size,
                              hipStream_t stream) {
    (void)in_sizes; (void)n_in; (void)out_size; (void)d_ws; (void)ws_size;
    const float* x   = (const float*)d_in[0];
    const float* cen = (const float*)d_in[1];
    int* out = (int*)d_out;
    // 1024 waves = B(4) * M/16(256); 4 waves per 128-thread block -> 256 blocks
    hipLaunchKernelGGL(knn_wmma_kernel, dim3(256), dim3(128), 0, stream, x, cen, out);
}